// SM_AP_RF_AttentionHead_35442070127262
// MI455X (gfx1250) — compile-verified
//
#include <hip/hip_runtime.h>
#include <hip/hip_bf16.h>

// Problem constants
#define BB 8
#define SS 2048
#define HH 768
#define DD 32

typedef __attribute__((ext_vector_type(2))) float v2f;
typedef __attribute__((ext_vector_type(8))) float v8f;

static __device__ __forceinline__ v8f wmma_f32_4(v2f a, v2f b, v8f c) {
    // D = A(16x4 f32) * B(4x16 f32) + C(16x16 f32)
    return __builtin_amdgcn_wmma_f32_16x16x4_f32(
        /*neg_a=*/false, a, /*neg_b=*/false, b,
        /*c_mod=*/(short)0, c, /*reuse_a=*/false, /*reuse_b=*/false);
}

// ---------------------------------------------------------------------------
// Kernel A: fold RF into projection weights.
//   Wqr[h][f] = sum_d Wq[d][h]*RF[d][f];  bqr[f] = sum_d bq[d]*RF[d][f]
//   WvT[h][f] = Wv[f][h]                  (uniform [H][32] layout for kernel B)
// ---------------------------------------------------------------------------
__global__ void fuse_weights_kernel(const float* __restrict__ Wq, const float* __restrict__ bq,
                                    const float* __restrict__ Wk, const float* __restrict__ bk,
                                    const float* __restrict__ Wv, const float* __restrict__ bv,
                                    const float* __restrict__ RF,
                                    float* __restrict__ Wqr, float* __restrict__ Wkr,
                                    float* __restrict__ WvT,
                                    float* __restrict__ bqr, float* __restrict__ bkr,
                                    float* __restrict__ bvc) {
    const int h = blockIdx.x;   // 0..767
    const int f = threadIdx.x;  // 0..31
    float aq = 0.f, ak = 0.f;
    for (int d = 0; d < DD; ++d) {
        const float rf = RF[d * DD + f];
        aq += Wq[d * HH + h] * rf;
        ak += Wk[d * HH + h] * rf;
    }
    Wqr[h * DD + f] = aq;
    Wkr[h * DD + f] = ak;
    WvT[h * DD + f] = Wv[f * HH + h];
    if (h == 0) {
        float sq = 0.f, sk = 0.f;
        for (int d = 0; d < DD; ++d) {
            const float rf = RF[d * DD + f];
            sq += bq[d] * rf;
            sk += bk[d] * rf;
        }
        bqr[f] = sq;
        bkr[f] = sk;
        bvc[f] = bv[f];
    }
}

// ---------------------------------------------------------------------------
// Kernel B: projections. X[16384 x 768] * W[768 x 32] -> {qr, kr, v}.
// One 16-row slab per block; 6 waves = 3 outputs x 2 N-tiles.
// X chunk (16x64) staged in LDS; 192 f32-WMMAs per wave.
// ---------------------------------------------------------------------------
__global__ void proj_kernel(const float* __restrict__ x,
                            const float* __restrict__ Wqr, const float* __restrict__ Wkr,
                            const float* __restrict__ WvT,
                            const float* __restrict__ bqr, const float* __restrict__ bkr,
                            const float* __restrict__ bvc,
                            float* __restrict__ qr, float* __restrict__ kr,
                            float* __restrict__ vv) {
    __shared__ float Xs[16][64];
    const int tid  = threadIdx.x;   // 0..191
    const int lane = tid & 31;
    const int wv   = tid >> 5;      // 0..5
    const int o    = wv >> 1;       // 0=qr 1=kr 2=v
    const int nt   = wv & 1;        // N-tile
    const int m0   = blockIdx.x * 16;

    const float* W    = (o == 0) ? Wqr : (o == 1) ? Wkr : WvT;
    const float* bias = (o == 0) ? bqr : (o == 1) ? bkr : bvc;
    float*       outp = (o == 0) ? qr  : (o == 1) ? kr  : vv;

    const int row  = lane & 15;           // A-matrix M / B-matrix N lane role
    const int kh   = (lane >> 4) * 2;     // K sub-offset per ISA f32 layout
    const int ncol = nt * 16 + row;

    v8f acc = {};
    for (int kc = 0; kc < HH / 64; ++kc) {
        for (int idx = tid; idx < 16 * 64; idx += 192) {
            const int r = idx >> 6, c = idx & 63;
            Xs[r][c] = x[(size_t)(m0 + r) * HH + kc * 64 + c];
        }
        __syncthreads();
#pragma unroll
        for (int k4 = 0; k4 < 16; ++k4) {
            v2f a, bfr;
            a.x = Xs[row][k4 * 4 + kh];
            a.y = Xs[row][k4 * 4 + kh + 1];
            const int kg = kc * 64 + k4 * 4 + kh;
            bfr.x = W[(size_t)kg * DD + ncol];
            bfr.y = W[(size_t)(kg + 1) * DD + ncol];
            acc = wmma_f32_4(a, bfr, acc);
        }
        __syncthreads();
    }
    const float bval = bias[ncol];
#pragma unroll
    for (int j = 0; j < 8; ++j) {
        const int rw = j + 8 * (lane >> 4);
        outp[(size_t)(m0 + rw) * DD + ncol] = acc[j] + bval;
    }
}

// ---------------------------------------------------------------------------
// Kernel C: raw scores = qr * kr^T / sqrt(D), written into the probs region.
// Block = 4 waves -> 16 q-rows x 64 t-cols. qr/kr slabs are L2-resident.
// ---------------------------------------------------------------------------
__global__ void scores_kernel(const float* __restrict__ qr, const float* __restrict__ kr,
                              float* __restrict__ probs) {
    const int tid  = threadIdx.x;
    const int lane = tid & 31;
    const int wv   = tid >> 5;                 // 0..3
    const int b    = blockIdx.z;
    const int s0   = blockIdx.x * 16;
    const int t0   = (blockIdx.y * 4 + wv) * 16;
    const int ln   = lane & 15;
    const int kh   = (lane >> 4) * 2;
    const size_t qbase = ((size_t)b * SS + s0) * DD;
    const size_t kbase = ((size_t)b * SS + t0) * DD;

    v8f acc = {};
#pragma unroll
    for (int j = 0; j < 8; ++j) {              // K = 32 in steps of 4
        v2f a, bb;
        a.x  = qr[qbase + (size_t)ln * DD + j * 4 + kh];
        a.y  = qr[qbase + (size_t)ln * DD + j * 4 + kh + 1];
        bb.x = kr[kbase + (size_t)ln * DD + j * 4 + kh];
        bb.y = kr[kbase + (size_t)ln * DD + j * 4 + kh + 1];
        acc = wmma_f32_4(a, bb, acc);
    }
    const float sc = 0.17677669529663687f;     // 1/sqrt(32)
#pragma unroll
    for (int j = 0; j < 8; ++j) {
        const int rw = j + 8 * (lane >> 4);
        probs[((size_t)b * SS + s0 + rw) * SS + t0 + ln] = acc[j] * sc;
    }
}

// ---------------------------------------------------------------------------
// Kernel D: fused softmax + AV for a 16-query slab.
//   P1: rowmax/rowsum over raw scores (coalesced float4 scans)
//   P2: out = softmax(scores) @ V via WMMA, exp/normalize applied to A-frags
//       on the fly; K=2048 split over wave pairs, partials summed in LDS.
//   P3: rewrite probs normalized (coalesced float4 RMW).
// ---------------------------------------------------------------------------
__global__ void softmax_av_kernel(const float* __restrict__ vv,
                                  float* __restrict__ probs, float* __restrict__ out) {
    __shared__ float red[16][8];
    __shared__ float rowmax[16];
    __shared__ float rowscale[16];
    __shared__ float cpart[2][8][32];

    const int tid  = threadIdx.x;   // 0..127
    const int lane = tid & 31;
    const int wv   = tid >> 5;      // 0..3
    const int b    = blockIdx.y;
    const int s0   = blockIdx.x * 16;
    float* praw = probs + ((size_t)b * SS + s0) * SS;

    // Phase 1a: row max
    {
        const int r = tid >> 3, sub = tid & 7;
        float m = -3.4e38f;
        for (int i = 0; i < 64; ++i) {
            const float4 q = *(const float4*)(praw + (size_t)r * SS + (size_t)(i * 8 + sub) * 4);
            m = fmaxf(m, fmaxf(fmaxf(q.x, q.y), fmaxf(q.z, q.w)));
        }
        red[r][sub] = m;
    }
    __syncthreads();
    if (tid < 16) {
        float m = red[tid][0];
        for (int j = 1; j < 8; ++j) m = fmaxf(m, red[tid][j]);
        rowmax[tid] = m;
    }
    __syncthreads();
    // Phase 1b: row sum of exp
    {
        const int r = tid >> 3, sub = tid & 7;
        const float rm = rowmax[r];
        float s = 0.f;
        for (int i = 0; i < 64; ++i) {
            const float4 q = *(const float4*)(praw + (size_t)r * SS + (size_t)(i * 8 + sub) * 4);
            s += __expf(q.x - rm) + __expf(q.y - rm) + __expf(q.z - rm) + __expf(q.w - rm);
        }
        red[r][sub] = s;
    }
    __syncthreads();
    if (tid < 16) {
        float s = 0.f;
        for (int j = 0; j < 8; ++j) s += red[tid][j];
        rowscale[tid] = 1.0f / s;
    }
    __syncthreads();

    // Phase 2: AV GEMM with on-the-fly softmax on A-fragments
    const int nt    = wv & 1;        // v column tile
    const int khalf = wv >> 1;       // K range half
    const int row   = lane & 15;
    const int kh    = (lane >> 4) * 2;
    const int ncol  = nt * 16 + row;
    const float rm  = rowmax[row];
    const float rs  = rowscale[row];

    v8f acc = {};
    for (int k4 = 0; k4 < 256; ++k4) {
        const int kb = khalf * 1024 + k4 * 4;
        v2f a, bb;
        const float r0 = praw[(size_t)row * SS + kb + kh];
        const float r1 = praw[(size_t)row * SS + kb + kh + 1];
        a.x = __expf(r0 - rm) * rs;
        a.y = __expf(r1 - rm) * rs;
        bb.x = vv[((size_t)b * SS + kb + kh) * DD + ncol];
        bb.y = vv[((size_t)b * SS + kb + kh + 1) * DD + ncol];
        acc = wmma_f32_4(a, bb, acc);
    }
    __syncthreads();   // all raw-score reads complete before phase 3 rewrites
    if (khalf == 1) {
#pragma unroll
        for (int j = 0; j < 8; ++j) cpart[nt][j][lane] = acc[j];
    }
    __syncthreads();
    if (khalf == 0) {
#pragma unroll
        for (int j = 0; j < 8; ++j) {
            const int rw = j + 8 * (lane >> 4);
            out[((size_t)b * SS + s0 + rw) * DD + ncol] = acc[j] + cpart[nt][j][lane];
        }
    }

    // Phase 3: normalize probs in place (coalesced)
    {
        const int r = tid >> 3, sub = tid & 7;
        const float rm2 = rowmax[r], rs2 = rowscale[r];
        for (int i = 0; i < 64; ++i) {
            float4* p = (float4*)(praw + (size_t)r * SS + (size_t)(i * 8 + sub) * 4);
            float4 q = *p;
            q.x = __expf(q.x - rm2) * rs2;
            q.y = __expf(q.y - rm2) * rs2;
            q.z = __expf(q.z - rm2) * rs2;
            q.w = __expf(q.w - rm2) * rs2;
            *p = q;
        }
    }
}

extern "C" void kernel_launch(void* const* d_in, const int* in_sizes, int n_in,
                              void* d_out, int out_size, void* d_ws, size_t ws_size,
                              hipStream_t stream) {
    const float* x  = (const float*)d_in[0];
    const float* Wq = (const float*)d_in[1];
    const float* bq = (const float*)d_in[2];
    const float* Wk = (const float*)d_in[3];
    const float* bk = (const float*)d_in[4];
    const float* Wv = (const float*)d_in[5];
    const float* bv = (const float*)d_in[6];
    const float* RF = (const float*)d_in[7];

    float* out   = (float*)d_out;                       // [B,S,32]
    float* probs = out + (size_t)BB * SS * DD;          // [B,S,S]

    // Workspace layout (floats): ~6.6 MB total
    float* ws  = (float*)d_ws;
    float* qr  = ws;                                    // 524288
    float* kr  = qr  + (size_t)BB * SS * DD;            // 524288
    float* vv  = kr  + (size_t)BB * SS * DD;            // 524288
    float* Wqr = vv  + (size_t)BB * SS * DD;            // 24576
    float* Wkr = Wqr + (size_t)HH * DD;
    float* WvT = Wkr + (size_t)HH * DD;
    float* bqr = WvT + (size_t)HH * DD;
    float* bkr = bqr + DD;
    float* bvc = bkr + DD;

    fuse_weights_kernel<<<HH, DD, 0, stream>>>(Wq, bq, Wk, bk, Wv, bv, RF,
                                               Wqr, Wkr, WvT, bqr, bkr, bvc);
    proj_kernel<<<(BB * SS) / 16, 192, 0, stream>>>(x, Wqr, Wkr, WvT,
                                                    bqr, bkr, bvc, qr, kr, vv);
    scores_kernel<<<dim3(SS / 16, SS / 64, BB), 128, 0, stream>>>(qr, kr, probs);
    softmax_av_kernel<<<dim3(SS / 16, BB), 128, 0, stream>>>(vv, probs, out);
}